// MainGAT_61340722921802
// MI455X (gfx1250) — compile-verified
//
#include <hip/hip_runtime.h>
#include <math.h>

#define NN 50000
#define EE 800000
#define DD 64
#define EDD 16
#define LL 4
#define GG 64
#define FC_OUT 2048
#define NEG_SLOPE 0.2f

typedef float v2f __attribute__((ext_vector_type(2)));
typedef float v8f __attribute__((ext_vector_type(8)));

// ---------------------------------------------------------------------------
// Generic f32 WMMA GEMM:  C[M,N] = A[M,K] @ B[K,N] (+ bias[N])
// One wave computes one 16x16 tile via V_WMMA_F32_16X16X4_F32 (f32-exact).
// M, K, N multiples of 16 (50000=3125*16, 64, 2048). EXEC stays all-ones.
// ---------------------------------------------------------------------------
__global__ void wmma_gemm_f32_kernel(const float* __restrict__ A,
                                     const float* __restrict__ B,
                                     const float* __restrict__ bias,
                                     float* __restrict__ C,
                                     int M, int K, int N, int has_bias)
{
    const int lane = threadIdx.x & 31;
    const int wave = threadIdx.x >> 5;
    const int row0 = blockIdx.x * 16;
    const int col0 = (blockIdx.y * 4 + wave) * 16;
    const int half = lane >> 4;
    const int mr   = lane & 15;

    if (row0 >= M || col0 >= N) return;

    v8f c = {};
    const float* Arow = A + (size_t)(row0 + mr) * K;
    for (int k0 = 0; k0 < K; k0 += 4) {
        const int ka = k0 + 2 * half;
        v2f a, b;
        a.x = Arow[ka];
        a.y = Arow[ka + 1];
        b.x = B[(size_t)ka * N + col0 + mr];
        b.y = B[(size_t)(ka + 1) * N + col0 + mr];
        c = __builtin_amdgcn_wmma_f32_16x16x4_f32(
            false, a, false, b, (short)0, c, false, false);
    }
    const float bb = has_bias ? bias[col0 + mr] : 0.0f;
#pragma unroll
    for (int r = 0; r < 8; ++r) {
        C[(size_t)(row0 + r + 8 * half) * N + col0 + mr] = c[r] + bb;
    }
}

// ---------------------------------------------------------------------------
// CSR construction (deterministic: int atomics + per-node sort)
// ---------------------------------------------------------------------------
__global__ void izero_kernel(int* __restrict__ p, int n) {
    int i = blockIdx.x * blockDim.x + threadIdx.x;
    if (i < n) p[i] = 0;
}

__global__ void count_deg_kernel(const int* __restrict__ dst, int* __restrict__ deg) {
    int e = blockIdx.x * blockDim.x + threadIdx.x;
    if (e < EE) atomicAdd(deg + dst[e], 1);
}

// single-block exclusive scan of deg[0..n) -> off[0..n], off[n] = total
__global__ void scan_kernel(const int* __restrict__ deg, int* __restrict__ off, int n) {
    __shared__ int buf[256];
    __shared__ int carry;
    if (threadIdx.x == 0) carry = 0;
    __syncthreads();
    for (int base = 0; base < n; base += 256) {
        int i = base + threadIdx.x;
        int v = (i < n) ? deg[i] : 0;
        buf[threadIdx.x] = v;
        __syncthreads();
        for (int s = 1; s < 256; s <<= 1) {
            int t = (threadIdx.x >= s) ? buf[threadIdx.x - s] : 0;
            __syncthreads();
            buf[threadIdx.x] += t;
            __syncthreads();
        }
        int incl = buf[threadIdx.x];
        if (i < n) off[i] = carry + incl - v;
        __syncthreads();
        if (threadIdx.x == 255) carry += buf[255];
        __syncthreads();
    }
    if (threadIdx.x == 0) off[n] = carry;
}

__global__ void icopy_kernel(const int* __restrict__ a, int* __restrict__ b, int n) {
    int i = blockIdx.x * blockDim.x + threadIdx.x;
    if (i < n) b[i] = a[i];
}

__global__ void fill_eid_kernel(const int* __restrict__ dst,
                                int* __restrict__ pos, int* __restrict__ eid) {
    int e = blockIdx.x * blockDim.x + threadIdx.x;
    if (e < EE) {
        int idx = atomicAdd(pos + dst[e], 1);
        eid[idx] = e;
    }
}

// restore deterministic per-node ordering (lists are ~16 long)
__global__ void sort_eid_kernel(const int* __restrict__ off, int* __restrict__ eid) {
    int n = blockIdx.x * blockDim.x + threadIdx.x;
    if (n >= NN) return;
    int s = off[n], e = off[n + 1];
    for (int i = s + 1; i < e; ++i) {
        int v = eid[i];
        int j = i - 1;
        while (j >= s && eid[j] > v) { eid[j + 1] = eid[j]; --j; }
        eid[j + 1] = v;
    }
}

// batch is sorted -> graphs contiguous; tiny scan of G=64 counts
__global__ void count_batch_kernel(const int* __restrict__ batch, int* __restrict__ cntG) {
    int n = blockIdx.x * blockDim.x + threadIdx.x;
    if (n < NN) atomicAdd(cntG + batch[n], 1);
}

__global__ void scan_g_kernel(const int* __restrict__ cntG, int* __restrict__ offG) {
    if (threadIdx.x == 0) {
        int acc = 0;
        for (int g = 0; g < GG; ++g) { offG[g] = acc; acc += cntG[g]; }
        offG[GG] = acc;
    }
}

// ---------------------------------------------------------------------------
// Per-layer pieces
// ---------------------------------------------------------------------------
__global__ void wea_kernel(const float* __restrict__ We,
                           const float* __restrict__ a_e,
                           float* __restrict__ w_ea) {
    int j = threadIdx.x;
    if (j < EDD) {
        float s = 0.0f;
        for (int d = 0; d < DD; ++d) s += We[j * DD + d] * a_e[d];
        w_ea[j] = s;
    }
}

__global__ void node_scores_kernel(const float* __restrict__ hw,
                                   const float* __restrict__ a_s,
                                   const float* __restrict__ a_d,
                                   float* __restrict__ ssrc,
                                   float* __restrict__ sdst) {
    int n = blockIdx.x * blockDim.x + threadIdx.x;
    if (n >= NN) return;
    const float* row = hw + (size_t)n * DD;
    float s0 = 0.0f, s1 = 0.0f;
#pragma unroll 8
    for (int d = 0; d < DD; ++d) {
        float h = row[d];
        s0 += h * a_s[d];
        s1 += h * a_d[d];
    }
    ssrc[n] = s0;
    sdst[n] = s1;
}

// score[e] = leaky_relu(s_src[src] + s_dst[dst] + edge_attr[e].w_ea)
__global__ void edge_score_kernel(const int* __restrict__ src,
                                  const int* __restrict__ dst,
                                  const float* __restrict__ edge_attr,
                                  const float* __restrict__ w_ea,
                                  const float* __restrict__ ssrc,
                                  const float* __restrict__ sdst,
                                  float* __restrict__ score) {
    int e = blockIdx.x * blockDim.x + threadIdx.x;
    if (e >= EE) return;
    const float* ea = edge_attr + (size_t)e * EDD;
    float el = 0.0f;
#pragma unroll
    for (int j = 0; j < EDD; ++j) el += ea[j] * w_ea[j];
    float sc = ssrc[src[e]] + sdst[dst[e]] + el;
    score[e] = (sc >= 0.0f) ? sc : NEG_SLOPE * sc;
}

// One wave per dst node: segment-max, softmax denom, weighted gather of
// hw[src] rows, bias (+optional relu). No float atomics; single store per row.
__global__ void gat_aggregate_kernel(const int* __restrict__ off,
                                     const int* __restrict__ eid,
                                     const int* __restrict__ src,
                                     const float* __restrict__ score,
                                     const float* __restrict__ hw,
                                     const float* __restrict__ b,
                                     float* __restrict__ out,
                                     int apply_relu) {
    int node = blockIdx.x * (blockDim.x >> 5) + (threadIdx.x >> 5);
    int lane = threadIdx.x & 31;
    if (node >= NN) return;
    int s = off[node], e = off[node + 1];

    float acc0 = 0.0f, acc1 = 0.0f;
    if (e > s) {
        // segment max (lane-strided, butterfly reduce)
        float mx = -INFINITY;
        for (int i = s + lane; i < e; i += 32) mx = fmaxf(mx, score[eid[i]]);
#pragma unroll
        for (int o = 16; o; o >>= 1) mx = fmaxf(mx, __shfl_xor(mx, o, 32));
        // softmax denominator
        float ds = 0.0f;
        for (int i = s + lane; i < e; i += 32) ds += expf(score[eid[i]] - mx);
#pragma unroll
        for (int o = 16; o; o >>= 1) ds += __shfl_xor(ds, o, 32);
        float inv = 1.0f / ds;
        // weighted feature gather: lane owns features {lane, lane+32}
        for (int i = s; i < e; ++i) {
            int ee = eid[i];
            float alpha = expf(score[ee] - mx) * inv;
            const float* row = hw + (size_t)src[ee] * DD;
            acc0 += alpha * row[lane];
            acc1 += alpha * row[lane + 32];
        }
    }
    float o0 = acc0 + b[lane];
    float o1 = acc1 + b[lane + 32];
    if (apply_relu) { o0 = fmaxf(o0, 0.0f); o1 = fmaxf(o1, 0.0f); }
    out[(size_t)node * DD + lane]      = o0;
    out[(size_t)node * DD + lane + 32] = o1;
}

// per-graph mean pool (graphs contiguous since batch is sorted)
__global__ void pool_gather_kernel(const int* __restrict__ offG,
                                   const float* __restrict__ h,
                                   float* __restrict__ pooled) {
    int g = blockIdx.x;
    int f = threadIdx.x;              // 64 features
    int s = offG[g], e = offG[g + 1];
    float acc = 0.0f;
    for (int n = s; n < e; ++n) acc += h[(size_t)n * DD + f];
    pooled[(size_t)g * DD + f] = acc / fmaxf((float)(e - s), 1.0f);
}

// ---------------------------------------------------------------------------
static inline int ceil_div(long long a, int b) { return (int)((a + b - 1) / b); }

extern "C" void kernel_launch(void* const* d_in, const int* in_sizes, int n_in,
                              void* d_out, int out_size, void* d_ws, size_t ws_size,
                              hipStream_t stream) {
    (void)in_sizes; (void)n_in; (void)out_size; (void)ws_size;

    const float* x         = (const float*)d_in[0];
    const int*   eidx      = (const int*)  d_in[1];
    const float* edge_attr = (const float*)d_in[2];
    const int*   batch     = (const int*)  d_in[3];
    const float* Wall      = (const float*)d_in[4];
    const float* a_src_all = (const float*)d_in[5];
    const float* a_dst_all = (const float*)d_in[6];
    const float* We_all    = (const float*)d_in[7];
    const float* a_e_all   = (const float*)d_in[8];
    const float* b_all     = (const float*)d_in[9];
    const float* Wfc       = (const float*)d_in[10];
    const float* bfc       = (const float*)d_in[11];
    float*       outp      = (float*)d_out;

    const int* src = eidx;
    const int* dst = eidx + EE;

    // workspace carve-up
    float* ws = (float*)d_ws;
    size_t off_f = 0;
    float* hw     = ws + off_f; off_f += (size_t)NN * DD;
    float* hA     = ws + off_f; off_f += (size_t)NN * DD;
    float* ssrc   = ws + off_f; off_f += NN;
    float* sdst   = ws + off_f; off_f += NN;
    float* score  = ws + off_f; off_f += EE;
    float* w_ea   = ws + off_f; off_f += EDD;
    float* pooled = ws + off_f; off_f += (size_t)GG * DD;
    int* iws  = (int*)(ws + off_f);
    size_t off_i = 0;
    int* deg  = iws + off_i; off_i += NN;
    int* offs = iws + off_i; off_i += NN + 1;
    int* pos  = iws + off_i; off_i += NN;
    int* eid  = iws + off_i; off_i += EE;
    int* cntG = iws + off_i; off_i += GG;
    int* offG = iws + off_i; off_i += GG + 1;

    const int T = 256;

    // ---- CSR over dst (once per call, deterministic) ----
    izero_kernel<<<ceil_div(NN, T), T, 0, stream>>>(deg, NN);
    count_deg_kernel<<<ceil_div(EE, T), T, 0, stream>>>(dst, deg);
    scan_kernel<<<1, 256, 0, stream>>>(deg, offs, NN);
    icopy_kernel<<<ceil_div(NN, T), T, 0, stream>>>(offs, pos, NN);
    fill_eid_kernel<<<ceil_div(EE, T), T, 0, stream>>>(dst, pos, eid);
    sort_eid_kernel<<<ceil_div(NN, T), T, 0, stream>>>(offs, eid);

    // ---- graph ranges for mean pool ----
    izero_kernel<<<1, T, 0, stream>>>(cntG, GG);
    count_batch_kernel<<<ceil_div(NN, T), T, 0, stream>>>(batch, cntG);
    scan_g_kernel<<<1, 32, 0, stream>>>(cntG, offG);

    // ---- GAT layers ----
    for (int l = 0; l < LL; ++l) {
        const float* h   = (l == 0) ? x : hA;
        const float* W   = Wall      + (size_t)l * DD * DD;
        const float* a_s = a_src_all + (size_t)l * DD;
        const float* a_d = a_dst_all + (size_t)l * DD;
        const float* We  = We_all    + (size_t)l * EDD * DD;
        const float* a_e = a_e_all   + (size_t)l * DD;
        const float* b   = b_all     + (size_t)l * DD;

        wmma_gemm_f32_kernel<<<dim3(NN / 16, DD / 64), 128, 0, stream>>>(
            h, W, nullptr, hw, NN, DD, DD, 0);
        node_scores_kernel<<<ceil_div(NN, T), T, 0, stream>>>(hw, a_s, a_d, ssrc, sdst);
        wea_kernel<<<1, 32, 0, stream>>>(We, a_e, w_ea);
        edge_score_kernel<<<ceil_div(EE, T), T, 0, stream>>>(
            src, dst, edge_attr, w_ea, ssrc, sdst, score);
        gat_aggregate_kernel<<<ceil_div(NN, 8), 256, 0, stream>>>(
            offs, eid, src, score, hw, b, hA, (l < LL - 1) ? 1 : 0);
    }

    // ---- mean pool + final FC (WMMA) ----
    pool_gather_kernel<<<GG, DD, 0, stream>>>(offG, hA, pooled);
    wmma_gemm_f32_kernel<<<dim3(GG / 16, FC_OUT / 64), 128, 0, stream>>>(
        pooled, Wfc, bfc, outp, GG, DD, FC_OUT, 1);
}